// TransformerPrior_86397562127149
// MI455X (gfx1250) — compile-verified
//
#include <hip/hip_runtime.h>
#include <math.h>

// Problem constants from the reference (fixed shapes).
constexpr int B = 32;
constexpr int T = 1024;
constexpr int S = 512;
constexpr int C = 384;

#define NEG_INF_F (-1.0e9f)
#define LOG_2PI_F 1.8378770664093453f

typedef __attribute__((ext_vector_type(2))) float v2f;
typedef __attribute__((ext_vector_type(8))) float v8f;

__device__ __forceinline__ float logaddexp_f(float a, float b) {
    float m = fmaxf(a, b);
    float n = fminf(a, b);
    // m + log1p(exp(n-m)); exact match for logaddexp incl. the -1e9 sentinel case
    return m + log1pf(__expf(n - m));
}

// ---------------------------------------------------------------------------
// K1: scores[b,t,s] = sum_c eps[b,t,c] * cond[b,s,c]
// One wave per 16x64 output strip: 4 independent accumulators share one A
// fragment per K-step (4x A reuse, 4 independent v_wmma chains to hide the
// multi-cycle f32-WMMA + load latency).
// A (16x4 f32) layout: lanes 0-15 -> M=lane, K={0,1}; lanes 16-31 -> M=lane-16, K={2,3}
// B (4x16 f32) layout mirrors A with N in place of M.
// C/D (16x16 f32): VGPR r -> (M=r, N=lane) lanes 0-15 ; (M=r+8, N=lane-16) lanes 16-31
// ---------------------------------------------------------------------------
constexpr int NT = 4;  // 16-wide S-tiles per wave

__global__ void __launch_bounds__(256)
gemm_scores_wmma(const float* __restrict__ eps,
                 const float* __restrict__ cond,
                 float* __restrict__ scores) {
    const int lane = threadIdx.x & 31;
    const int wave = (blockIdx.x * blockDim.x + threadIdx.x) >> 5;

    constexpr int stripsS   = S / (16 * NT);          // 8
    constexpr int perBatch  = (T / 16) * stripsS;     // 64*8 = 512
    const int b   = wave / perBatch;
    const int rem = wave % perBatch;
    const int t0  = (rem / stripsS) * 16;
    const int s0  = (rem % stripsS) * (16 * NT);

    const int mn   = lane & 15;                       // row (A) / col (B)
    const int kOff = (lane >> 4) << 1;                // 0 or 2

    const float* Abase = eps  + (size_t)b * T * C + (size_t)(t0 + mn) * C + kOff;
    const float* Bbase = cond + (size_t)b * S * C + (size_t)(s0 + mn) * C + kOff;

    v8f acc[NT];
#pragma unroll
    for (int n = 0; n < NT; ++n) acc[n] = (v8f){0.f,0.f,0.f,0.f,0.f,0.f,0.f,0.f};

#pragma unroll 4
    for (int k = 0; k < C; k += 4) {
        const v2f a = *(const v2f*)(Abase + k);
        v2f bf[NT];
#pragma unroll
        for (int n = 0; n < NT; ++n)
            bf[n] = *(const v2f*)(Bbase + (size_t)n * 16 * C + k);
#pragma unroll
        for (int n = 0; n < NT; ++n) {
            // (neg_a, A, neg_b, B, c_mod, C, reuse_a, reuse_b)
            acc[n] = __builtin_amdgcn_wmma_f32_16x16x4_f32(
                false, a, false, bf[n], (short)0, acc[n], false, false);
        }
    }

    float* out = scores + (size_t)b * T * S;
    const int rowBase = t0 + ((lane >> 4) << 3);      // +0 or +8
#pragma unroll
    for (int n = 0; n < NT; ++n) {
        const int col = s0 + n * 16 + mn;
#pragma unroll
        for (int r = 0; r < 8; ++r)
            out[(size_t)(rowBase + r) * S + col] = acc[n][r];
    }
}

// ---------------------------------------------------------------------------
// K2: softmax over axis T (per (b,s) column), in place. Coalesced across s.
// ---------------------------------------------------------------------------
__global__ void __launch_bounds__(256)
softmax_over_t(float* __restrict__ scores) {
    const int b = blockIdx.y;
    const int s = blockIdx.x * blockDim.x + threadIdx.x;
    if (s >= S) return;
    float* base = scores + (size_t)b * T * S + s;

    float m = -INFINITY;
    for (int t = 0; t < T; ++t) m = fmaxf(m, base[(size_t)t * S]);
    float sum = 0.f;
    for (int t = 0; t < T; ++t) sum += __expf(base[(size_t)t * S] - m);
    const float inv = 1.0f / sum;
    for (int t = 0; t < T; ++t)
        base[(size_t)t * S] = __expf(base[(size_t)t * S] - m) * inv;
}

// ---------------------------------------------------------------------------
// K3: base log-prob per batch: sum_{t,c} -0.5*(eps^2 + log(2pi))
// ---------------------------------------------------------------------------
__global__ void __launch_bounds__(256)
base_logprob_kernel(const float* __restrict__ eps, float* __restrict__ baseLp) {
    const int b = blockIdx.x;
    const float* p = eps + (size_t)b * T * C;
    float acc = 0.f;
    for (int i = threadIdx.x; i < T * C; i += blockDim.x) {
        float v = p[i];
        acc += v * v;
    }
    __shared__ float red[256];
    red[threadIdx.x] = acc;
    __syncthreads();
    for (int off = 128; off > 0; off >>= 1) {
        if (threadIdx.x < off) red[threadIdx.x] += red[threadIdx.x + off];
        __syncthreads();
    }
    if (threadIdx.x == 0)
        baseLp[b] = -0.5f * (red[0] + (float)(T * C) * LOG_2PI_F);
}

// ---------------------------------------------------------------------------
// K4: backward DP (beta), one workgroup per batch, S threads, prev row in LDS.
// beta[T-1,j] = (j==S-1 ? 0 : -inf)
// beta[t,j]   = logaddexp(beta[t+1,j]+E[t+1,j], beta[t+1,j+1]+E[t+1,j+1])
// global_prefetch of the next row shortens the serial per-step critical path.
// ---------------------------------------------------------------------------
__global__ void __launch_bounds__(S)
backward_beta(const float* __restrict__ P, float* __restrict__ beta) {
    const int b = blockIdx.x;
    const int j = threadIdx.x;
    const float* Pb = P + (size_t)b * T * S;
    float* Bb = beta + (size_t)b * T * S;

    __shared__ float lm[S];

    float bj = (j == S - 1) ? 0.f : NEG_INF_F;
    Bb[(size_t)(T - 1) * S + j] = bj;

    for (int t = T - 2; t >= 0; --t) {
        const float e = Pb[(size_t)(t + 1) * S + j];
        if (t > 0)  // pull next step's row toward the WGP while we sync/compute
            __builtin_prefetch(Pb + (size_t)t * S + j, 0, 1);
        lm[j] = bj + e;
        __syncthreads();
        float right = (j == S - 1) ? NEG_INF_F : lm[j + 1];
        bj = logaddexp_f(lm[j], right);
        Bb[(size_t)t * S + j] = bj;
        __syncthreads();
    }
}

// ---------------------------------------------------------------------------
// K5: fused forward DP (mu) + Viterbi move extraction.
// Viterbi on logpi = mu + beta - mu_N is invariant to the constant mu_N
// (it shifts all candidates at time t equally), so we run it on mu_t+beta_t
// and never materialize mu or logpi. Emits moves[b,t,j] (t>=1) and mu_N[b].
// ---------------------------------------------------------------------------
__global__ void __launch_bounds__(S)
forward_viterbi(const float* __restrict__ P,
                const float* __restrict__ beta,
                unsigned char* __restrict__ moves,
                float* __restrict__ muN) {
    const int b = blockIdx.x;
    const int j = threadIdx.x;
    const float* Pb = P + (size_t)b * T * S;
    const float* Bb = beta + (size_t)b * T * S;
    unsigned char* Mb = moves + (size_t)b * T * S;

    __shared__ float lmu[S];
    __shared__ float lv[S];

    const float start = (j == 0) ? 0.f : NEG_INF_F;
    float mu = Pb[j] + start;                 // mu[0,j]
    float v  = mu + Bb[j] + start;            // v[0,j] (extra 'start' harmless: stays ~-inf)

    for (int t = 1; t < T; ++t) {
        const float e  = Pb[(size_t)t * S + j];
        const float bt = Bb[(size_t)t * S + j];
        if (t + 1 < T) {
            __builtin_prefetch(Pb + (size_t)(t + 1) * S + j, 0, 1);
            __builtin_prefetch(Bb + (size_t)(t + 1) * S + j, 0, 1);
        }
        lmu[j] = mu;
        lv[j]  = v;
        __syncthreads();
        const float muL = (j == 0) ? NEG_INF_F : lmu[j - 1];
        const float vL  = (j == 0) ? NEG_INF_F : lv[j - 1];
        const float vC  = lv[j];
        mu = e + logaddexp_f(lmu[j], muL);
        Mb[(size_t)t * S + j] = (vL > vC) ? 1 : 0;   // move = (shifted > v_prev)
        v = (mu + bt) + fmaxf(vC, vL);
        __syncthreads();
    }
    if (j == S - 1) muN[b] = mu;
}

// ---------------------------------------------------------------------------
// K6: serial backtrack (thread 0) + parallel one-hot path emission +
// sample_logprob = sum_t P[b,t,js[t]] - mu_N + base_logprob.
// ---------------------------------------------------------------------------
__global__ void __launch_bounds__(256)
backtrack_emit(const float* __restrict__ P,
               const unsigned char* __restrict__ moves,
               const float* __restrict__ baseLp,
               const float* __restrict__ muN,
               float* __restrict__ out) {
    const int b = blockIdx.x;
    __shared__ int js[T];

    if (threadIdx.x == 0) {
        const unsigned char* Mb = moves + (size_t)b * T * S;
        int j = S - 1;
        for (int t = T - 1; t >= 1; --t) {
            js[t] = j;
            j -= (int)Mb[(size_t)t * S + j];
        }
        js[0] = j;
    }
    __syncthreads();

    float* path = out + (size_t)b * T * S;
    const float* Pb = P + (size_t)b * T * S;
    float acc = 0.f;
    for (int idx = threadIdx.x; idx < T * S; idx += blockDim.x) {
        const int t = idx >> 9;          // /S (S=512)
        const int s = idx & (S - 1);
        const int hit = (s == js[t]);
        path[idx] = (float)hit;
        if (hit) acc += Pb[idx];
    }

    __shared__ float red[256];
    red[threadIdx.x] = acc;
    __syncthreads();
    for (int off = 128; off > 0; off >>= 1) {
        if (threadIdx.x < off) red[threadIdx.x] += red[threadIdx.x + off];
        __syncthreads();
    }
    if (threadIdx.x == 0)
        out[(size_t)B * T * S + b] = baseLp[b] + red[0] - muN[b];
}

// ---------------------------------------------------------------------------
extern "C" void kernel_launch(void* const* d_in, const int* in_sizes, int n_in,
                              void* d_out, int out_size, void* d_ws, size_t ws_size,
                              hipStream_t stream) {
    (void)in_sizes; (void)n_in; (void)out_size; (void)ws_size;

    const float* eps  = (const float*)d_in[0];   // [B,T,C]
    const float* cond = (const float*)d_in[1];   // [B,S,C]
    // d_in[2]/d_in[3] (lengths) are full-length in this problem; unused.

    float* out = (float*)d_out;                  // path [B,T,S] then logprobs [B]

    // Workspace layout
    const size_t NP = (size_t)B * T * S;         // 16,777,216
    float* P    = (float*)d_ws;                  // softmax(Wr) in place (= E)
    float* beta = P + NP;
    unsigned char* moves = (unsigned char*)(beta + NP);
    float* baseLp = (float*)(moves + NP);        // NP is 4-aligned
    float* muN    = baseLp + B;

    // K1: WMMA GEMM -> raw scores (one wave per 16x64 strip)
    {
        const int totalWaves = B * (T / 16) * (S / (16 * NT));  // 16384
        gemm_scores_wmma<<<totalWaves * 32 / 256, 256, 0, stream>>>(eps, cond, P);
    }
    // K2: softmax over T (in place)
    softmax_over_t<<<dim3(S / 256, B), 256, 0, stream>>>(P);
    // K3: base logprobs
    base_logprob_kernel<<<B, 256, 0, stream>>>(eps, baseLp);
    // K4: backward DP
    backward_beta<<<B, S, 0, stream>>>(P, beta);
    // K5: forward DP fused with Viterbi moves
    forward_viterbi<<<B, S, 0, stream>>>(P, beta, moves, muN);
    // K6: backtrack + path + logprobs
    backtrack_emit<<<B, 256, 0, stream>>>(P, moves, baseLp, muN, out);
}